// CARAFE_3453153706071
// MI455X (gfx1250) — compile-verified
//
#include <hip/hip_runtime.h>
#include <hip/hip_bf16.h>
#include <math.h>

typedef __attribute__((ext_vector_type(2))) float v2f;
typedef __attribute__((ext_vector_type(8))) float v8f;

// ---------------------------------------------------------------------------
// Generic row-major fp32 GEMM + bias using V_WMMA_F32_16X16X4_F32.
//   C[M,N] = A[M,K] * W[N,K]^T + bias[N]
// One wave computes a 16 x (16*NT) output tile. ISA lane layout (wave32):
//   A (16x4):  lane L, vgpr v -> A[L&15][2*(L>>4)+v]
//   B (4x16):  lane L, vgpr v -> B[2*(L>>4)+v][L&15]  (B[k][n] = W[n][k])
//   D (16x16): lane L, vgpr i -> D[i + 8*(L>>4)][L&15]
// ---------------------------------------------------------------------------
template <int NT>
__global__ void gemm_wmma(const float* __restrict__ A,
                          const float* __restrict__ W,
                          const float* __restrict__ bias,
                          float* __restrict__ C,
                          int M, int N, int K, int n_groups) {
  const int wid  = blockIdx.x * (blockDim.x >> 5) + (threadIdx.x >> 5);
  const int lane = threadIdx.x & 31;
  const int mt = wid / n_groups;
  const int ng = wid % n_groups;
  const int m0 = mt * 16;
  const int n0 = ng * (16 * NT);
  if (m0 >= M) return;  // wave-uniform

  const int r  = lane & 15;
  const int kh = lane >> 4;

  const float* Ap = A + (size_t)(m0 + r) * K + 2 * kh;
  const float* Wp = W + (size_t)(n0 + r) * K + 2 * kh;

  v8f acc[NT] = {};
  for (int k = 0; k < K; k += 4) {
    if ((k & 63) == 0) __builtin_prefetch(Ap + k + 64, 0, 1);
    v2f a = *(const v2f*)(Ap + k);
#pragma unroll
    for (int t = 0; t < NT; ++t) {
      v2f b = *(const v2f*)(Wp + (size_t)(16 * t) * K + k);
      acc[t] = __builtin_amdgcn_wmma_f32_16x16x4_f32(
          false, a, false, b, (short)0, acc[t], false, false);
    }
  }

  const int rowb = m0 + 8 * kh;
#pragma unroll
  for (int t = 0; t < NT; ++t) {
    const int col = n0 + 16 * t + (lane & 15);
    const float bv = bias[col];
#pragma unroll
    for (int i = 0; i < 8; ++i) {
      C[(size_t)(rowb + i) * N + col] = acc[t][i] + bv;
    }
  }
}

// ---------------------------------------------------------------------------
// Pad/transpose enc weights: Wpad[n][tap*64+c] = enc_w[n][c][di][dj] (n<36), 0 pad
// ---------------------------------------------------------------------------
__global__ void prep_wpad(const float* __restrict__ enc_w, float* __restrict__ Wpad) {
  const int idx = blockIdx.x * blockDim.x + threadIdx.x;  // 48*576 = 27648
  const int n = idx / 576;
  const int k = idx % 576;
  const int tap = k >> 6;
  const int c = k & 63;
  Wpad[idx] = (n < 36) ? enc_w[(size_t)(n * 64 + c) * 9 + tap] : 0.0f;
}

// ---------------------------------------------------------------------------
// enc 3x3 conv: on-the-fly im2col GEMM. M=65536, K=576, Npad=48 (36 valid).
// ---------------------------------------------------------------------------
__global__ void enc_conv_wmma(const float* __restrict__ kt1,
                              const float* __restrict__ Wpad,
                              const float* __restrict__ bias,
                              float* __restrict__ out36) {
  constexpr int K = 576, NT = 3;
  const int wid  = blockIdx.x * (blockDim.x >> 5) + (threadIdx.x >> 5);
  const int lane = threadIdx.x & 31;
  const int m0 = wid * 16;

  const int r  = lane & 15;
  const int kh = lane >> 4;
  const int m  = m0 + r;
  const int b  = m >> 12;
  const int hw = m & 4095;
  const int h  = hw >> 6;
  const int w  = hw & 63;

  const float* Wp = Wpad + (size_t)r * K + 2 * kh;

  v8f acc[NT] = {};
  for (int k = 0; k < K; k += 4) {
    const int kk  = k + 2 * kh;
    const int tap = kk >> 6;
    const int c   = kk & 63;
    const int hh  = h + tap / 3 - 1;
    const int ww  = w + tap % 3 - 1;
    v2f a;
    if ((unsigned)hh < 64u && (unsigned)ww < 64u) {
      a = *(const v2f*)(kt1 + (((size_t)b * 4096 + hh * 64 + ww) * 64 + c));
    } else {
      a.x = 0.0f; a.y = 0.0f;
    }
#pragma unroll
    for (int t = 0; t < NT; ++t) {
      v2f bb = *(const v2f*)(Wp + (size_t)(16 * t) * K + k);
      acc[t] = __builtin_amdgcn_wmma_f32_16x16x4_f32(
          false, a, false, bb, (short)0, acc[t], false, false);
    }
  }

  const int rowb = m0 + 8 * kh;
#pragma unroll
  for (int t = 0; t < NT; ++t) {
    const int col = 16 * t + (lane & 15);
    if (col < 36) {
      const float bv = bias[col];
#pragma unroll
      for (int i = 0; i < 8; ++i) {
        out36[(size_t)(rowb + i) * 36 + col] = acc[t][i] + bv;
      }
    }
  }
}

// ---------------------------------------------------------------------------
// pixel_shuffle(2) + softmax over K^2 + unfold rearrangement.
// enc36[bhw][k*4+u] -> ktsm[bhw][u*9+k] = softmax_k(...)
// ---------------------------------------------------------------------------
__global__ void softmax9(const float* __restrict__ enc36, float* __restrict__ ktsm) {
  const int t = blockIdx.x * blockDim.x + threadIdx.x;  // 262144
  const int u = t & 3;
  const size_t bhw = (size_t)(t >> 2);
  const float* e = enc36 + bhw * 36;
  float v[9];
  float mx = -3.4e38f;
#pragma unroll
  for (int k = 0; k < 9; ++k) {
    v[k] = e[4 * k + u];
    mx = fmaxf(mx, v[k]);
  }
  float s = 0.0f;
#pragma unroll
  for (int k = 0; k < 9; ++k) {
    v[k] = __expf(v[k] - mx);
    s += v[k];
  }
  const float inv = 1.0f / s;
  float* o = ktsm + bhw * 36 + u * 9;
#pragma unroll
  for (int k = 0; k < 9; ++k) o[k] = v[k] * inv;
}

// ---------------------------------------------------------------------------
// Fused reassembly + out 1x1 conv.
// Block = 256 threads (8 waves) owns 32 consecutive output rows
//   m = (b*128 + row2)*128 + col2,  col2 = c0..c0+31  (fixed b,row2).
// Phase 1: cooperative reassembly of the 32x256 feature tile into LDS
//   (row stride padded to 258 floats -> conflict-free ds_load_b64 A-fragments).
// Phase 2: 8 waves = 2 M-halves x 4 N-groups; 16x64 WMMA tiles, A from LDS,
//   B = out_w (L2-resident, 256 KB), bias epilogue -> d_out.
// ---------------------------------------------------------------------------
#define YLDS_STRIDE 258

__global__ void carafe_out_fused(const float* __restrict__ x,
                                 const float* __restrict__ ktsm,
                                 const float* __restrict__ out_w,
                                 const float* __restrict__ out_b,
                                 float* __restrict__ out) {
  __shared__ float ylds[32 * YLDS_STRIDE];  // 33,024 B

  const int block  = blockIdx.x;            // 8192
  const size_t m_base = (size_t)block * 32;
  const int brow = block >> 2;              // b*128 + row2
  const int b    = brow >> 7;
  const int row2 = brow & 127;
  const int h    = row2 >> 1;
  const int ri   = row2 & 1;
  const int c0   = (block & 3) * 32;        // first col2 of this block

  // ---- Phase 1: reassembly into LDS ----
  {
    const int c = threadIdx.x;  // channel 0..255
    const int w0 = c0 >> 1;
#pragma unroll 4
    for (int ww16 = 0; ww16 < 16; ++ww16) {
      const int w = w0 + ww16;
      float xv[9];
#pragma unroll
      for (int tap = 0; tap < 9; ++tap) {
        const int hh = h + tap / 3 - 1;
        const int ww = w + tap % 3 - 1;
        xv[tap] = ((unsigned)hh < 64u && (unsigned)ww < 64u)
                      ? x[((size_t)b * 4096 + hh * 64 + ww) * 256 + c]
                      : 0.0f;
      }
      const float* wtb = ktsm + ((size_t)b * 4096 + h * 64 + w) * 36 + ri * 18;
#pragma unroll
      for (int rj = 0; rj < 2; ++rj) {
        const float* wt = wtb + rj * 9;  // u = ri*2+rj, block-uniform -> s_load
        float acc = 0.0f;
#pragma unroll
        for (int k = 0; k < 9; ++k) acc += wt[k] * xv[k];
        ylds[(ww16 * 2 + rj) * YLDS_STRIDE + c] = acc;
      }
    }
  }
  __syncthreads();

  // ---- Phase 2: WMMA GEMM from LDS ----
  const int wv   = threadIdx.x >> 5;  // 0..7
  const int lane = threadIdx.x & 31;
  const int mtl  = wv >> 2;           // which 16-row half of the LDS tile
  const int ng   = wv & 3;            // 64-column group of N=256
  const int rsel = lane & 15;
  const int kh   = lane >> 4;

  const float* Wp = out_w + (size_t)(ng * 64 + rsel) * 256 + 2 * kh;
  const float* Al = ylds + (mtl * 16 + rsel) * YLDS_STRIDE + 2 * kh;

  v8f acc[4] = {};
  for (int k = 0; k < 256; k += 4) {
    v2f a = *(const v2f*)(Al + k);  // ds_load_b64 from LDS
#pragma unroll
    for (int t = 0; t < 4; ++t) {
      v2f bb = *(const v2f*)(Wp + (size_t)(16 * t) * 256 + k);
      acc[t] = __builtin_amdgcn_wmma_f32_16x16x4_f32(
          false, a, false, bb, (short)0, acc[t], false, false);
    }
  }

  const size_t rowb = m_base + (size_t)(mtl * 16 + 8 * kh);
#pragma unroll
  for (int t = 0; t < 4; ++t) {
    const int col = ng * 64 + 16 * t + (lane & 15);
    const float bv = out_b[col];
#pragma unroll
    for (int i = 0; i < 8; ++i) {
      out[(rowb + i) * 256 + col] = acc[t][i] + bv;
    }
  }
}

// ---------------------------------------------------------------------------
// Orchestration.
// ---------------------------------------------------------------------------
extern "C" void kernel_launch(void* const* d_in, const int* in_sizes, int n_in,
                              void* d_out, int out_size, void* d_ws, size_t ws_size,
                              hipStream_t stream) {
  const float* x      = (const float*)d_in[0];  // (16,4096,256)
  const float* down_w = (const float*)d_in[1];  // (64,256)
  const float* down_b = (const float*)d_in[2];  // (64)
  const float* enc_w  = (const float*)d_in[3];  // (36,64,3,3)
  const float* enc_b  = (const float*)d_in[4];  // (36)
  const float* out_w  = (const float*)d_in[5];  // (256,256)
  const float* out_b  = (const float*)d_in[6];  // (256)
  float* out = (float*)d_out;                   // (16,16384,256)

  float* ws   = (float*)d_ws;
  float* kt1  = ws;                   // 65536*64 = 4194304 f (16 MB)
  float* Wpad = kt1 + 4194304;        // 48*576   = 27648 f
  float* enc  = Wpad + 27648;         // 65536*36 = 2359296 f (9 MB)
  float* ktsm = enc + 2359296;        // 65536*36 = 2359296 f (9 MB)

  // 1) down 1x1 conv: (65536x256)*(256->64). 4096 waves, 8 waves/block.
  gemm_wmma<4><<<512, 256, 0, stream>>>(x, down_w, down_b, kt1,
                                        65536, 64, 256, /*n_groups=*/1);

  // 2) enc 3x3 conv as im2col GEMM, N padded to 48.
  prep_wpad<<<108, 256, 0, stream>>>(enc_w, Wpad);
  enc_conv_wmma<<<512, 256, 0, stream>>>(kt1, Wpad, enc_b, enc);

  // 3) pixel-shuffle + softmax over K^2.
  softmax9<<<1024, 256, 0, stream>>>(enc, ktsm);

  // 4) fused reassembly + out 1x1 conv -> d_out.
  carafe_out_fused<<<8192, 256, 0, stream>>>(x, ktsm, out_w, out_b, out);
}